// SeperatedBlockUpdateLayer_43714177138947
// MI455X (gfx1250) — compile-verified
//
#include <hip/hip_runtime.h>

typedef float v2f __attribute__((ext_vector_type(2)));
typedef float v8f __attribute__((ext_vector_type(8)));

#define BN_EPS 1e-5f

// Ragged block geometry (compile-time constants from the reference).
__device__ __forceinline__ int block_of_row(int row)
{
    if (row < 131072) return 0;
    if (row < 278528) return 1;
    if (row < 409600) return 2;
    if (row < 512000) return 3;
    return 4;
}
__constant__ float c_invCnt[5] = {1.0f / 131072.0f, 1.0f / 147456.0f, 1.0f / 131072.0f,
                                  1.0f / 102400.0f, 1.0f / 165888.0f};

// ---------------------------------------------------------------------------
// Stage A: per-channel block matmul  h[b,i,k,c] = sum_j x[b,i,j,c]*x[b,j,k,c]
// One workgroup per batch element; the batch tile (n*n*64 floats, <=64KB) is a
// contiguous chunk of x, staged through LDS.
// ---------------------------------------------------------------------------
__global__ __launch_bounds__(256) void blockmm_kernel(const float* __restrict__ x,
                                                      float* __restrict__ out)
{
    const int cumB[6]   = {0, 2048, 3072, 3584, 4608, 6656};
    const int nsz[5]    = {8, 12, 16, 10, 9};
    const int rowOff[5] = {0, 131072, 278528, 409600, 512000};

    int wg = blockIdx.x;
    int bid = 0;
#pragma unroll
    for (int t = 1; t < 5; ++t)
        if (wg >= cumB[t]) bid = t;
    int batch   = wg - cumB[bid];
    int n       = nsz[bid];
    int rowBase = rowOff[bid] + batch * n * n;
    int total   = n * n * 64;

    extern __shared__ float tile[];     // up to 16*16*64 floats = 64 KB
    const float* src = x + (size_t)rowBase * 64;
    for (int idx = threadIdx.x; idx < total; idx += 256) tile[idx] = src[idx];
    __syncthreads();

    for (int idx = threadIdx.x; idx < total; idx += 256) {
        int c  = idx & 63;          // consecutive lanes -> consecutive LDS banks
        int ik = idx >> 6;
        int i  = ik / n;
        int k  = ik - i * n;
        float acc = 0.f;
        for (int j = 0; j < n; ++j)
            acc += tile[(i * n + j) * 64 + c] * tile[(j * n + k) * 64 + c];
        out[(size_t)(rowBase + ik) * 64 + c] = acc;
    }
}

// ---------------------------------------------------------------------------
// One K-segment of the WMMA GEMM: pointers pre-offset per lane, advanced by
// +4 floats per step (no per-iteration address rebuilds or source selects).
// BN-on-load resolved at compile time via template parameter.
// ---------------------------------------------------------------------------
template <bool BN>
__device__ __forceinline__ void gemm_kseg(
    v8f acc[4],
    const float* __restrict__ ap,        // A row base + 2*laneHi
    const float* __restrict__ wp_base,   // W + laneLo*K + kOff + 2*laneHi
    int K, int segK,
    const float* __restrict__ scp,       // LDS bnsc + kOff + 2*laneHi
    const float* __restrict__ shp)       // LDS bnsh + kOff + 2*laneHi
{
    for (int kk = 0; kk < segK; kk += 4) {
        v2f a = *(const v2f*)ap;
        if (BN) {                        // fused BN + ReLU of producer stage
            v2f sc = *(const v2f*)scp;
            v2f sh = *(const v2f*)shp;
            a.x = fmaxf(a.x * sc.x + sh.x, 0.f);
            a.y = fmaxf(a.y * sc.y + sh.y, 0.f);
            scp += 4;
            shp += 4;
        }
        const float* wp = wp_base;
#pragma unroll
        for (int t = 0; t < 4; ++t) {
            v2f b = *(const v2f*)wp;
            acc[t] = __builtin_amdgcn_wmma_f32_16x16x4_f32(
                false, a, false, b, (short)0, acc[t], false, false);
            wp += 16 * K;                // next 16-column output tile
        }
        ap += 4;
        wp_base += 4;
    }
}

// ---------------------------------------------------------------------------
// Fused GEMM: out(N x 64) = relu(bn([A0|A1])) @ W(64 x K)^T + bias
//   - Optional per-column BN+ReLU applied to A on load (producer's raw sums ->
//     per-channel scale/shift precomputed into LDS once per workgroup).
//   - Per-output-channel sum/sumsq of the (bias-added) result accumulated via
//     shfl_xor(16) + LDS ds_add_f32 + global float atomics for the next BN.
// Wave = 16 rows x 64 out-cols via V_WMMA_F32_16X16X4_F32, K stepped by 4.
// fp32 A-frag layout (ISA 7.12.2): lanes 0-15 hold K={kk,kk+1}, lanes 16-31
// hold K={kk+2,kk+3}; C/D: VGPR r -> row r (lanes<16) / row r+8 (lanes>=16),
// column = lane&15.  Workgroup tile = 128 contiguous rows; all ragged-block
// boundaries are multiples of 128 so the BN block id is workgroup-uniform.
// ---------------------------------------------------------------------------
__global__ __launch_bounds__(256) void gemm_wmma_fused_kernel(
    const float* __restrict__ A0, const float* __restrict__ A1,
    int KA0, int K,
    const float* __restrict__ W,    // (64, K) row-major
    const float* __restrict__ bias, // (64)
    // input BN for A0 columns [0,KA0): raw sums slot(s), gamma, beta
    const float* __restrict__ st0, const float* __restrict__ gam0,
    const float* __restrict__ bet0, int perBlk0, float invCnt0,
    // input BN for A1 columns [KA0,K)
    const float* __restrict__ st1, const float* __restrict__ gam1,
    const float* __restrict__ bet1, int perBlk1, float invCnt1,
    // output stats accumulation (raw sum/sumsq of out incl. bias)
    float* __restrict__ ostats, int operBlk,
    float* __restrict__ out, int nrows)
{
    __shared__ float bnsc[128], bnsh[128];  // per-A-column scale/shift
    __shared__ float osum[64], osq[64];     // per-out-channel partial stats

    int tid    = threadIdx.x;
    int lane   = tid & 31;
    int wave   = tid >> 5;
    int laneLo = lane & 15;
    int laneHi = lane >> 4;
    int wgRow  = blockIdx.x * 128;
    int bid    = block_of_row(wgRow);

    bool bn0 = (st0 != nullptr);
    bool bn1 = (st1 != nullptr);

    // --- setup: per-channel BN scale/shift tables + zero out-stats ---
    if (tid < 64) { osum[tid] = 0.f; osq[tid] = 0.f; }
    if (tid < K) {
        float sc = 1.f, sh = 0.f;
        bool on;
        const float* st; const float* gg; const float* bb; float ic; int c;
        if (tid < KA0) { on = bn0; st = st0; gg = gam0; bb = bet0;
                         ic = perBlk0 ? c_invCnt[bid] : invCnt0;
                         if (on && perBlk0) st += bid * 128; c = tid; }
        else           { on = bn1; st = st1; gg = gam1; bb = bet1;
                         ic = perBlk1 ? c_invCnt[bid] : invCnt1;
                         if (on && perBlk1) st += bid * 128; c = tid - KA0; }
        if (on) {
            float mean = st[c] * ic;
            float var  = st[64 + c] * ic - mean * mean;
            sc = gg[c] * rsqrtf(var + BN_EPS);
            sh = bb[c] - mean * sc;
        }
        bnsc[tid] = sc;
        bnsh[tid] = sh;
    }
    __syncthreads();

    int row0 = wgRow + wave * 16;
    if (row0 < nrows) {
        v8f acc[4] = {};                // 4 x (16x16) tiles = 64 output channels
        int aRow  = row0 + laneLo;
        int khalf = 2 * laneHi;

        // Segment 1: columns [0, KA0) from A0
        {
            const float* ap = A0 + (size_t)aRow * KA0 + khalf;
            const float* wp = W + (size_t)laneLo * K + khalf;
            if (bn0) gemm_kseg<true >(acc, ap, wp, K, KA0, bnsc + khalf, bnsh + khalf);
            else     gemm_kseg<false>(acc, ap, wp, K, KA0, nullptr, nullptr);
        }
        // Segment 2: columns [KA0, K) from A1
        if (K > KA0) {
            int K1 = K - KA0;
            const float* ap = A1 + (size_t)aRow * K1 + khalf;
            const float* wp = W + (size_t)laneLo * K + KA0 + khalf;
            if (bn1) gemm_kseg<true >(acc, ap, wp, K, K1, bnsc + KA0 + khalf, bnsh + KA0 + khalf);
            else     gemm_kseg<false>(acc, ap, wp, K, K1, nullptr, nullptr);
        }

#pragma unroll
        for (int t = 0; t < 4; ++t) {
            int ocol = t * 16 + laneLo;
            float bs = bias[ocol];
            float s = 0.f, q = 0.f;
#pragma unroll
            for (int r = 0; r < 8; ++r) {
                int row = row0 + r + 8 * laneHi;
                float v = acc[t][r] + bs;
                out[(size_t)row * 64 + ocol] = v;
                s += v;
                q += v * v;
            }
            // combine the two lane-halves holding the same output column
            s += __shfl_xor(s, 16, 32);
            q += __shfl_xor(q, 16, 32);
            if (laneHi == 0) {
                atomicAdd(&osum[ocol], s);   // ds_add_f32
                atomicAdd(&osq[ocol], q);
            }
        }
    }
    __syncthreads();

    // --- flush workgroup stats to global (128 atomics per WG) ---
    if (tid < 64) {
        float* slot = operBlk ? (ostats + bid * 128) : ostats;
        atomicAdd(&slot[tid], osum[tid]);
        atomicAdd(&slot[64 + tid], osq[tid]);
    }
}

__global__ void zero_kernel(float* __restrict__ p, int n)
{
    int i = blockIdx.x * blockDim.x + threadIdx.x;
    if (i < n) p[i] = 0.f;
}

// ---------------------------------------------------------------------------
// Final: out = relu(bn(A; stats,gamma,beta)) + x   (global BN over all rows)
// ---------------------------------------------------------------------------
__global__ __launch_bounds__(256) void bn_relu_add_kernel(
    const float* __restrict__ A, int rowCount,
    const float* __restrict__ stats, const float* __restrict__ gamma,
    const float* __restrict__ beta, float invCnt,
    const float* __restrict__ addX, float* __restrict__ dst)
{
    size_t total = (size_t)rowCount * 64;
    for (size_t idx = (size_t)blockIdx.x * 256 + threadIdx.x; idx < total;
         idx += (size_t)gridDim.x * 256) {
        int c      = (int)(idx & 63);
        float mean = stats[c] * invCnt;
        float var  = stats[64 + c] * invCnt - mean * mean;
        float sc   = gamma[c] * rsqrtf(var + BN_EPS);
        float v    = (A[idx] - mean) * sc + beta[c];
        v = fmaxf(v, 0.f) + addX[idx];
        dst[idx] = v;
    }
}

// ---------------------------------------------------------------------------
extern "C" void kernel_launch(void* const* d_in, const int* in_sizes, int n_in,
                              void* d_out, int out_size, void* d_ws, size_t ws_size,
                              hipStream_t stream)
{
    const float* x   = (const float*)d_in[0];
    const float* cw  = (const float*)d_in[1];   // (2,64,64)
    const float* cb  = (const float*)d_in[2];   // (2,64)
    const float* cg  = (const float*)d_in[3];
    const float* cbe = (const float*)d_in[4];
    const float* w1  = (const float*)d_in[5];   // (64,128)
    const float* b1  = (const float*)d_in[6];
    const float* g1  = (const float*)d_in[7];
    const float* be1 = (const float*)d_in[8];
    const float* w2  = (const float*)d_in[9];   // (64,64)
    const float* b2  = (const float*)d_in[10];
    const float* g2  = (const float*)d_in[11];
    const float* be2 = (const float*)d_in[12];

    const int N = 677888;                       // total rows
    float* P = (float*)d_out;                   // N x 64 ping buffer (final output)
    float* Q = (float*)d_ws;                    // N x 64 pong buffer
    float* stats = Q + (size_t)N * 64;          // stage stats (raw sum/sumsq)
    float* stC0 = stats;                        // conv d=0: 5 blocks x 128
    float* stC1 = stats + 640;                  // conv d=1: 5 blocks x 128
    float* stM1 = stats + 1280;                 // mlp1: 128
    float* stM2 = stats + 1408;                 // mlp2: 128  (total 1536)

    const int gemmGrid = N / 128;               // 8 waves x 16 rows per WG
    const int ewGrid   = (int)(((size_t)N * 64 + 255) / 256);
    const float invN   = 1.0f / (float)N;

    // zero all stats slots once per call (deterministic, capture-safe)
    zero_kernel<<<6, 256, 0, stream>>>(stats, 1536);

    // Stage A: h0 = per-channel block matmul  -> Q (no BN on input of conv0)
    blockmm_kernel<<<6656, 256, 64 * 1024, stream>>>(x, Q);

    // Conv d=0: P = Q @ cw0^T + cb0 ; accumulate per-block stats -> stC0
    gemm_wmma_fused_kernel<<<gemmGrid, 256, 0, stream>>>(
        Q, Q, 64, 64, cw, cb,
        nullptr, nullptr, nullptr, 0, 0.f,
        nullptr, nullptr, nullptr, 0, 0.f,
        stC0, 1, P, N);

    // Conv d=1: Q = relu(bn(P; stC0,cg0,cbe0)) @ cw1^T + cb1 ; stats -> stC1
    gemm_wmma_fused_kernel<<<gemmGrid, 256, 0, stream>>>(
        P, P, 64, 64, cw + 4096, cb + 64,
        stC0, cg, cbe, 1, 0.f,
        nullptr, nullptr, nullptr, 0, 0.f,
        stC1, 1, Q, N);

    // MLP1: P = [x | relu(bn(Q; stC1,cg1,cbe1))] @ w1^T + b1 ; stats -> stM1
    gemm_wmma_fused_kernel<<<gemmGrid, 256, 0, stream>>>(
        x, Q, 64, 128, w1, b1,
        nullptr, nullptr, nullptr, 0, 0.f,
        stC1, cg + 64, cbe + 64, 1, 0.f,
        stM1, 0, P, N);

    // MLP2: Q = relu(bn(P; stM1,g1,be1)) @ w2^T + b2 ; stats -> stM2
    gemm_wmma_fused_kernel<<<gemmGrid, 256, 0, stream>>>(
        P, P, 64, 64, w2, b2,
        stM1, g1, be1, 0, invN,
        nullptr, nullptr, nullptr, 0, 0.f,
        stM2, 0, Q, N);

    // Final: d_out = relu(bn(Q; stM2,g2,be2)) + x
    bn_relu_add_kernel<<<ewGrid, 256, 0, stream>>>(Q, N, stM2, g2, be2, invN, x, P);

    (void)in_sizes; (void)n_in; (void)out_size; (void)ws_size;
}